// Graph_Conv_32598801777294
// MI455X (gfx1250) — compile-verified
//
#include <hip/hip_runtime.h>

typedef __attribute__((ext_vector_type(2))) float v2f;
typedef __attribute__((ext_vector_type(8))) float v8f;

#define BATCH 32
#define NN    1024
#define IND   128
#define HIDD  64
#define LDS_STRIDE 66   // 64 + 2 pad -> conflict-free b64 A-frag reads

// ---------------------------------------------------------------------------
// Kernel 1: H = x@W_h + b_h ; P = x@W_fc1 + b_fc1
// x viewed as [32768, 128]; one wave computes a 16x64 tile of BOTH outputs.
// grid = 256 blocks * 8 waves = 2048 waves = 32768/16 row tiles.
// ---------------------------------------------------------------------------
__global__ __launch_bounds__(256) void k_proj(
    const float* __restrict__ x,
    const float* __restrict__ Wh, const float* __restrict__ bh,
    const float* __restrict__ Wf, const float* __restrict__ bf,
    float* __restrict__ H, float* __restrict__ P) {
  const int lane = threadIdx.x & 31;
  const int w    = threadIdx.x >> 5;
  const int gw   = blockIdx.x * 8 + w;   // 0..2047
  const int m0   = gw * 16;              // row-tile base
  const int lr   = lane & 15;            // M (A) / N (B,C) index
  const int hi   = lane >> 4;            // K-pair selector

  v8f ha[4] = {}; v8f pa[4] = {};
  const float* xrow = x + (size_t)(m0 + lr) * IND + 2 * hi;

#pragma unroll 4
  for (int k0 = 0; k0 < IND; k0 += 4) {
    v2f a = *(const v2f*)(xrow + k0);                  // A frag: rows of x
    const float* w0 = Wh + (size_t)(k0 + 2 * hi) * HIDD + lr;
    const float* f0 = Wf + (size_t)(k0 + 2 * hi) * HIDD + lr;
#pragma unroll
    for (int nb = 0; nb < 4; ++nb) {
      v2f bw, bq;
      bw.x = w0[nb * 16]; bw.y = w0[HIDD + nb * 16];   // B frag: W_h cols
      bq.x = f0[nb * 16]; bq.y = f0[HIDD + nb * 16];   // B frag: W_fc1 cols
      ha[nb] = __builtin_amdgcn_wmma_f32_16x16x4_f32(false, a, false, bw,
                                                     (short)0, ha[nb], false, false);
      pa[nb] = __builtin_amdgcn_wmma_f32_16x16x4_f32(false, a, false, bq,
                                                     (short)0, pa[nb], false, false);
    }
  }

#pragma unroll
  for (int nb = 0; nb < 4; ++nb) {
    const int n = nb * 16 + lr;
    const float bhn = bh[n];
    const float bfn = bf[n];
#pragma unroll
    for (int i = 0; i < 8; ++i) {
      const int m = i + 8 * hi;                        // C layout: VGPR i, lane-half
      const size_t off = (size_t)(m0 + m) * HIDD + n;
      H[off] = ha[nb][i] + bhn;
      P[off] = pa[nb][i] + bfn;
    }
  }
}

// ---------------------------------------------------------------------------
// Kernel 2: x_new = A[b] @ H[b]  (K = 1024, WMMA f32 16x16x4)
// then fused gate: out = xn*sig(xn@Wg2 + P@Wg1 + bg1+bg2) + P*(1-sig)
// One wave owns a 16x64 output tile; 8 waves/block; grid = 32*64/8 = 256.
// ---------------------------------------------------------------------------
__global__ __launch_bounds__(256) void k_bmm_gate(
    const float* __restrict__ A, const float* __restrict__ H,
    const float* __restrict__ P,
    const float* __restrict__ Wg1, const float* __restrict__ bg1,
    const float* __restrict__ Wg2, const float* __restrict__ bg2,
    float* __restrict__ out) {
  __shared__ float lds[8 * 16 * LDS_STRIDE];           // per-wave 16x64 xn tile (padded)

  const int lane = threadIdx.x & 31;
  const int w    = threadIdx.x >> 5;
  const int gw   = blockIdx.x * 8 + w;                 // 0..2047
  const int b    = gw >> 6;                            // batch
  const int m0   = (gw & 63) * 16;                     // row tile in N
  const int lr   = lane & 15;
  const int hi   = lane >> 4;

  const float* Ab = A + (size_t)b * NN * NN;
  const float* Hb = H + (size_t)b * NN * HIDD;

  // ---- main bmm: acc = A[b][m0:m0+16, :] @ H[b]  -------------------------
  v8f acc[4] = {};
  const float* arow = Ab + (size_t)(m0 + lr) * NN + 2 * hi;
#pragma unroll 4
  for (int k0 = 0; k0 < NN; k0 += 4) {
    v2f a = *(const v2f*)(arow + k0);                  // sequential per-lane: full line use
    const float* h0 = Hb + (size_t)(k0 + 2 * hi) * HIDD + lr;
#pragma unroll
    for (int nb = 0; nb < 4; ++nb) {
      v2f bb;
      bb.x = h0[nb * 16];
      bb.y = h0[HIDD + nb * 16];
      acc[nb] = __builtin_amdgcn_wmma_f32_16x16x4_f32(false, a, false, bb,
                                                      (short)0, acc[nb], false, false);
    }
  }

  // ---- stage xn tile to LDS so we can re-read it as A-fragments ----------
  float* myl = lds + w * (16 * LDS_STRIDE);
#pragma unroll
  for (int nb = 0; nb < 4; ++nb)
#pragma unroll
    for (int i = 0; i < 8; ++i)
      myl[(i + 8 * hi) * LDS_STRIDE + nb * 16 + lr] = acc[nb][i];
  // same-wave producer/consumer through LDS: DScnt waits inserted by compiler

  // ---- g = xn@Wg2 + P_tile@Wg1  (both via WMMA, shared accumulator) ------
  v8f g[4] = {};
  const float* prow = P + (size_t)(b * NN + m0 + lr) * HIDD + 2 * hi;
#pragma unroll
  for (int kk = 0; kk < HIDD; kk += 4) {
    v2f a2 = *(const v2f*)(myl + lr * LDS_STRIDE + kk + 2 * hi);  // xn A-frag (LDS)
    v2f a1 = *(const v2f*)(prow + kk);                            // P  A-frag (global)
    const float* g2w = Wg2 + (size_t)(kk + 2 * hi) * HIDD + lr;
    const float* g1w = Wg1 + (size_t)(kk + 2 * hi) * HIDD + lr;
#pragma unroll
    for (int nb = 0; nb < 4; ++nb) {
      v2f b2, b1;
      b2.x = g2w[nb * 16]; b2.y = g2w[HIDD + nb * 16];
      b1.x = g1w[nb * 16]; b1.y = g1w[HIDD + nb * 16];
      g[nb] = __builtin_amdgcn_wmma_f32_16x16x4_f32(false, a2, false, b2,
                                                    (short)0, g[nb], false, false);
      g[nb] = __builtin_amdgcn_wmma_f32_16x16x4_f32(false, a1, false, b1,
                                                    (short)0, g[nb], false, false);
    }
  }

  // ---- fused sigmoid gate + store ----------------------------------------
#pragma unroll
  for (int nb = 0; nb < 4; ++nb) {
    const int n = nb * 16 + lr;
    const float bias = bg1[n] + bg2[n];
#pragma unroll
    for (int i = 0; i < 8; ++i) {
      const int m = i + 8 * hi;
      const size_t off = (size_t)(b * NN + m0 + m) * HIDD + n;
      const float xn = acc[nb][i];
      const float p  = P[off];
      const float z  = g[nb][i] + bias;
      const float gate = 1.0f / (1.0f + __expf(-z));
      out[off] = xn * gate + p * (1.0f - gate);
    }
  }
}

// ---------------------------------------------------------------------------
extern "C" void kernel_launch(void* const* d_in, const int* in_sizes, int n_in,
                              void* d_out, int out_size, void* d_ws, size_t ws_size,
                              hipStream_t stream) {
  const float* x    = (const float*)d_in[0];
  const float* A    = (const float*)d_in[1];
  const float* W_h  = (const float*)d_in[2];
  const float* b_h  = (const float*)d_in[3];
  const float* W_f1 = (const float*)d_in[4];
  const float* b_f1 = (const float*)d_in[5];
  const float* W_g1 = (const float*)d_in[6];
  const float* b_g1 = (const float*)d_in[7];
  const float* W_g2 = (const float*)d_in[8];
  const float* b_g2 = (const float*)d_in[9];
  float* out = (float*)d_out;

  float* H = (float*)d_ws;                         // [32768, 64] = 8 MB
  float* P = H + (size_t)BATCH * NN * HIDD;        // [32768, 64] = 8 MB

  k_proj<<<256, 256, 0, stream>>>(x, W_h, b_h, W_f1, b_f1, H, P);
  k_bmm_gate<<<256, 256, 0, stream>>>(A, H, P, W_g1, b_g1, W_g2, b_g2, out);
}